// NetVLADpp_76553497084250
// MI455X (gfx1250) — compile-verified
//
#include <hip/hip_runtime.h>

// ---------------------------------------------------------------------------
// NetVLAD++ for MI455X (gfx1250).
//
// Math collapse: out[bt,k,d] = alpha[bt,k] * (x[bt,d] - c[k,d]) where
//   logits = x @ Wt + b ; assn = softmax_K(logits)
//   s_k    = ||x||^2 - 2 x.c_k + ||c_k||^2        (= ||x - c_k||^2)
//   n_k    = assn_k * sqrt(s_k);  dn_k = max(n_k, EPS)
//   N      = sqrt( sum_k (n_k/dn_k)^2 );  alpha_k = assn_k / (dn_k * max(N,EPS))
//
// Phase 1 (tiny): per 16-timestep tile compute logits and x.c via
//   V_WMMA_F32_16X16X4_F32 (fp32 matrix core), then softmax + alpha -> ws.
// Phase 2 (512 MiB of stores, HBM-bound ~22us @ 23.3 TB/s): pure stream,
//   nontemporal float4 stores; x/centers re-reads hit the 192 MB L2.
// ---------------------------------------------------------------------------

typedef __attribute__((ext_vector_type(2))) float v2f;
typedef __attribute__((ext_vector_type(4))) float v4f;
typedef __attribute__((ext_vector_type(8))) float v8f;

#define TILE_T     16
#define D_DIM      512
#define K_CL       64
#define XS_STRIDE  520          // 512 + 8 floats: breaks 64-bank conflicts, keeps 8B align
#define NVLAD_EPS  1e-12f

// -------------------------- Phase 1: alpha table ---------------------------
__global__ __launch_bounds__(128)
void netvlad_alpha_kernel(const float* __restrict__ x,
                          const float* __restrict__ centers,
                          const float* __restrict__ conv_w,
                          const float* __restrict__ conv_b,
                          float* __restrict__ alpha)
{
    __shared__ float Xs[TILE_T * XS_STRIDE];          // 16 x 512 fp32 X tile (padded)
    __shared__ float Lg[TILE_T][K_CL + 8];            // logits, then exp(), then assn/dn
    __shared__ float Xc[TILE_T][K_CL + 8];            // x . c_k
    __shared__ float cn2[K_CL];                       // ||c_k||^2
    __shared__ float xn2[TILE_T];                     // ||x_t||^2

    const int tid = threadIdx.x;
    const int bt0 = blockIdx.x * TILE_T;              // first flattened (b*T + t)
    const float* __restrict__ xbase = x + (size_t)bt0 * D_DIM;

    // Stage the 16x512 X tile into LDS (coalesced float4).
    for (int i = tid; i < TILE_T * (D_DIM / 4); i += 128) {
        const int r  = i >> 7;                        // / (D_DIM/4)
        const int c4 = i & 127;
        v4f v = *(const v4f*)(xbase + (size_t)r * D_DIM + (size_t)c4 * 4);
        float* dst = &Xs[r * XS_STRIDE + c4 * 4];
        dst[0] = v.x; dst[1] = v.y; dst[2] = v.z; dst[3] = v.w;
    }
    __syncthreads();

    // ---- dual GEMM-let via fp32 WMMA: logits tile and x.c tile (16x64) ----
    // 4 waves, wave w owns output columns [16w, 16w+16).
    const int wave = tid >> 5;
    const int lane = tid & 31;
    const int mrow = lane & 15;                       // A: M = lane%16
    const int koff = (lane >> 4) << 1;                // A: VGPR0 holds K=koff, VGPR1 K=koff+1
    const int ncol = wave * 16 + mrow;                // B: N = lane%16 within this wave's tile

    const float* __restrict__ wrow = conv_w  + (size_t)ncol * D_DIM + koff;
    const float* __restrict__ crow = centers + (size_t)ncol * D_DIM + koff;
    const float* __restrict__ arow = &Xs[mrow * XS_STRIDE + koff];

    v8f accL = {};                                    // logits accumulator (16x16 f32)
    v8f accC = {};                                    // x.c accumulator
    for (int k = 0; k < D_DIM; k += 4) {
        v2f a;  a.x = arow[k];  a.y = arow[k + 1];    // ds_load_b64 (padded stride)
        v2f bw = *(const v2f*)(wrow + k);             // global float2, L2-resident
        v2f bc = *(const v2f*)(crow + k);
        accL = __builtin_amdgcn_wmma_f32_16x16x4_f32(
                   false, a, false, bw, (short)0, accL, false, false);
        accC = __builtin_amdgcn_wmma_f32_16x16x4_f32(
                   false, a, false, bc, (short)0, accC, false, false);
    }

    // D layout: VGPR j -> row M = j + 8*(lane/16), col N = lane%16.
    {
        const int mb  = (lane >> 4) << 3;
        const int col = wave * 16 + (lane & 15);
#pragma unroll
        for (int j = 0; j < 8; ++j) {
            Lg[mb + j][col] = accL[j];
            Xc[mb + j][col] = accC[j];
        }
    }

    // ||c_k||^2 (redundant per WG: 64x512 MACs, negligible) and ||x_t||^2.
    if (tid < K_CL) {
        const float* cr = centers + (size_t)tid * D_DIM;
        float s = 0.f;
        for (int d = 0; d < D_DIM; d += 4) {
            v4f v = *(const v4f*)(cr + d);
            s += v.x * v.x + v.y * v.y + v.z * v.z + v.w * v.w;
        }
        cn2[tid] = s;
    }
    if (tid < TILE_T) {
        const float* xr = &Xs[tid * XS_STRIDE];
        float s = 0.f;
        for (int d = 0; d < D_DIM; ++d) s += xr[d] * xr[d];
        xn2[tid] = s;
    }
    __syncthreads();

    // Per-timestep scalar epilogue: softmax over K, norms, alpha.
    if (tid < TILE_T) {
        const int m = tid;
        float mx = -3.0e38f;
        for (int k = 0; k < K_CL; ++k) {
            float l = Lg[m][k] + conv_b[k];
            Lg[m][k] = l;
            mx = fmaxf(mx, l);
        }
        float sum = 0.f;
        for (int k = 0; k < K_CL; ++k) {
            float e = __expf(Lg[m][k] - mx);
            Lg[m][k] = e;
            sum += e;
        }
        const float inv = 1.f / sum;
        const float x2  = xn2[m];
        float Nsq = 0.f;
        for (int k = 0; k < K_CL; ++k) {
            float assn = Lg[m][k] * inv;
            float s  = x2 - 2.f * Xc[m][k] + cn2[k];
            s = fmaxf(s, 0.f);
            float n  = assn * __fsqrt_rn(s);
            float dn = fmaxf(n, NVLAD_EPS);
            float r  = n / dn;
            Nsq += r * r;
            Lg[m][k] = assn / dn;                     // alpha before 2nd normalize
        }
        const float Ninv = 1.f / fmaxf(__fsqrt_rn(Nsq), NVLAD_EPS);
        float* __restrict__ adst = alpha + (size_t)(bt0 + m) * K_CL;
        for (int k = 0; k < K_CL; ++k) adst[k] = Lg[m][k] * Ninv;
    }
}

// ----------------------- Phase 2: HBM-rate streaming -----------------------
// out[btk, d] = alpha[btk] * (x[bt, d] - c[k, d]); 4 coalesced float4 / thread.
__global__ __launch_bounds__(256)
void netvlad_store_kernel(const float* __restrict__ x,
                          const float* __restrict__ centers,
                          const float* __restrict__ alpha,
                          float* __restrict__ out)
{
    const size_t base = (size_t)blockIdx.x * 1024 + threadIdx.x;
#pragma unroll
    for (int j = 0; j < 4; ++j) {
        const size_t idx = base + (size_t)j * 256;    // float4 index
        const size_t btk = idx >> 7;                  // D/4 = 128 float4 per row
        const int    d4  = (int)(idx & 127);
        const size_t bt  = btk >> 6;                  // K = 64
        const int    k   = (int)(btk & 63);

        const float a = alpha[btk];
        v4f xv = *(const v4f*)(x       + bt * (size_t)D_DIM + (size_t)d4 * 4);
        v4f cv = *(const v4f*)(centers + (size_t)k * D_DIM  + (size_t)d4 * 4);
        v4f o;
        o.x = a * (xv.x - cv.x);
        o.y = a * (xv.y - cv.y);
        o.z = a * (xv.z - cv.z);
        o.w = a * (xv.w - cv.w);
        // Output (512 MiB) >> L2: keep it out of cache with a nontemporal store.
        __builtin_nontemporal_store(o, (v4f*)out + idx);
    }
}

// ------------------------------- launcher ----------------------------------
extern "C" void kernel_launch(void* const* d_in, const int* in_sizes, int n_in,
                              void* d_out, int out_size, void* d_ws, size_t ws_size,
                              hipStream_t stream)
{
    const float* x       = (const float*)d_in[0];   // (B,T,D)
    const float* centers = (const float*)d_in[1];   // (K,D)
    const float* conv_w  = (const float*)d_in[2];   // (K,D)
    const float* conv_b  = (const float*)d_in[3];   // (K,)
    float* alpha = (float*)d_ws;                    // (B*T, K) = 1 MiB scratch
    float* out   = (float*)d_out;                   // (B,T,K*D)

    const int BT = in_sizes[0] / D_DIM;             // B*T = 4096

    netvlad_alpha_kernel<<<BT / TILE_T, 128, 0, stream>>>(
        x, centers, conv_w, conv_b, alpha);

    // total float4 = BT*K*D/4 = BT*8192 ; 1024 float4 per block
    netvlad_store_kernel<<<BT * 8, 256, 0, stream>>>(x, centers, alpha, out);
}